// CWG_30794915512765
// MI455X (gfx1250) — compile-verified
//
#include <hip/hip_runtime.h>
#include <cstdint>
#include <cstddef>

// Problem geometry (fixed by the reference).
#define BATCH   64
#define CH      256
#define SIZE_S  4096

// Tiling: 256 threads x 4 floats = 1024-wide s-tile per block.
#define THREADS 256
#define VEC     4
#define TILE_S  (THREADS * VEC)     // 1024
#define CSTEP   8                   // channel rows per pipeline stage
#define NSTAGES (CH / CSTEP)        // 32

// ---- CDNA5 async Global->LDS path (gfx1250), with portability guards ----
#if defined(__AMDGCN__) && __has_builtin(__builtin_amdgcn_global_load_async_to_lds_b128)
#define HAVE_ASYNC_LDS 1
typedef int v4i __attribute__((ext_vector_type(4)));
// builtin signature (inferred from diagnostics):
//   (AS1 v4i* src, AS3 v4i* dst, imm int offset, imm int cpol)
typedef __attribute__((address_space(1))) v4i* gbl_v4i_ptr;
typedef __attribute__((address_space(3))) v4i* lds_v4i_ptr;
#else
#define HAVE_ASYNC_LDS 0
#endif

#if defined(__AMDGCN__) && __has_builtin(__builtin_amdgcn_s_wait_asynccnt)
#define WAIT_ASYNC(n) __builtin_amdgcn_s_wait_asynccnt(n)
#elif defined(__AMDGCN__)
#define WAIT_ASYNC(n) asm volatile("s_wait_asynccnt %0" ::"i"(n) : "memory")
#else
#define WAIT_ASYNC(n) ((void)0)
#endif

__device__ __forceinline__ float capped_relu(float v) {
    return (v > 0.0f && v <= 1.0f) ? v : 0.0f;
}

__global__ __launch_bounds__(THREADS) void cdot_capped_relu_kernel(
    const float* __restrict__ x,      // [BATCH, CH, SIZE_S]
    const float* __restrict__ w,      // [CH, SIZE_S]
    const float* __restrict__ bias,   // [SIZE_S]
    float* __restrict__ out)          // [BATCH, SIZE_S]
{
#if HAVE_ASYNC_LDS
    __shared__ float xs[2][CSTEP][TILE_S];   // 64 KB double buffer
#endif
    const int tid = threadIdx.x;
    const int b   = blockIdx.y;
    const int s   = blockIdx.x * TILE_S + tid * VEC;   // 16B-aligned per lane

    const float* xrow = x + (size_t)b * CH * SIZE_S + s;  // stride SIZE_S per c
    float4 acc = make_float4(0.f, 0.f, 0.f, 0.f);

#if HAVE_ASYNC_LDS
    // Prologue: kick off stage 0 (8 async b128 loads; each lane targets its
    // own 16B LDS slot, so only this wave's ASYNCcnt gates consumption).
    {
#pragma unroll
        for (int j = 0; j < CSTEP; ++j) {
            __builtin_amdgcn_global_load_async_to_lds_b128(
                (gbl_v4i_ptr)(xrow + (size_t)j * SIZE_S),
                (lds_v4i_ptr)(&xs[0][j][tid * VEC]),
                /*offset=*/0, /*cpol=*/0);
        }
    }

    for (int stage = 0; stage < NSTAGES; ++stage) {
        const int buf = stage & 1;
        if (stage + 1 < NSTAGES) {
            const float* nsrc = xrow + (size_t)(stage + 1) * CSTEP * SIZE_S;
#pragma unroll
            for (int j = 0; j < CSTEP; ++j) {
                __builtin_amdgcn_global_load_async_to_lds_b128(
                    (gbl_v4i_ptr)(nsrc + (size_t)j * SIZE_S),
                    (lds_v4i_ptr)(&xs[buf ^ 1][j][tid * VEC]),
                    /*offset=*/0, /*cpol=*/0);
            }
            WAIT_ASYNC(CSTEP);   // this stage's 8 loads retired (in-order), next 8 in flight
        } else {
            WAIT_ASYNC(0);
        }
#pragma unroll
        for (int j = 0; j < CSTEP; ++j) {
            const int c = stage * CSTEP + j;
            const float4 wv = *reinterpret_cast<const float4*>(w + (size_t)c * SIZE_S + s);
            const float4 xv = *reinterpret_cast<const float4*>(&xs[buf][j][tid * VEC]);
            acc.x = fmaf(xv.x, wv.x, acc.x);
            acc.y = fmaf(xv.y, wv.y, acc.y);
            acc.z = fmaf(xv.z, wv.z, acc.z);
            acc.w = fmaf(xv.w, wv.w, acc.w);
        }
    }
#else
    // Fallback: plain register-streaming b128 loads.
#pragma unroll 8
    for (int c = 0; c < CH; ++c) {
        const float4 xv = *reinterpret_cast<const float4*>(xrow + (size_t)c * SIZE_S);
        const float4 wv = *reinterpret_cast<const float4*>(w + (size_t)c * SIZE_S + s);
        acc.x = fmaf(xv.x, wv.x, acc.x);
        acc.y = fmaf(xv.y, wv.y, acc.y);
        acc.z = fmaf(xv.z, wv.z, acc.z);
        acc.w = fmaf(xv.w, wv.w, acc.w);
    }
#endif

    const float4 bv = *reinterpret_cast<const float4*>(bias + s);
    float4 r;
    r.x = capped_relu(acc.x + bv.x);
    r.y = capped_relu(acc.y + bv.y);
    r.z = capped_relu(acc.z + bv.z);
    r.w = capped_relu(acc.w + bv.w);
    *reinterpret_cast<float4*>(out + (size_t)b * SIZE_S + s) = r;
}

extern "C" void kernel_launch(void* const* d_in, const int* in_sizes, int n_in,
                              void* d_out, int out_size, void* d_ws, size_t ws_size,
                              hipStream_t stream) {
    const float* x    = (const float*)d_in[0];   // [64, 256, 4096] f32
    const float* w    = (const float*)d_in[1];   // [256, 4096] f32
    const float* bias = (const float*)d_in[2];   // [4096] f32
    float* out        = (float*)d_out;           // [64, 4096] f32

    dim3 grid(SIZE_S / TILE_S, BATCH);           // (4, 64) = 256 blocks
    cdot_capped_relu_kernel<<<grid, THREADS, 0, stream>>>(x, w, bias, out);
}